// GraphPool_70806830842650
// MI455X (gfx1250) — compile-verified
//
#include <hip/hip_runtime.h>
#include <hip/hip_bf16.h>

// ---------------------------------------------------------------------------
// Problem: N=16384, IN_DIM=512, NUM_KEEP=8192
//   scores = tanh(|X @ w + b| / 100)        (16384x512 GEMV, trivial FLOPs)
//   idx    = top_k(scores, 8192) indices    (descending score, stable)
//   new_X  = X[idx, :]                      (17 MB gather)
//   A_new  = A[idx, :][:, idx]              (268 MB out, scattered reads)
// Memory-bound: ~0.5-0.8 GB traffic -> ~25-35 us at 23.3 TB/s.
// ---------------------------------------------------------------------------

typedef __attribute__((ext_vector_type(2))) float v2f;
typedef __attribute__((ext_vector_type(8))) float v8f;

#define NTOT      16384
#define IN_DIM    512
#define NUM_KEEP  8192

// ---------------------------------------------------------------------------
// Kernel 1: scores + sortable keys via V_WMMA_F32_16X16X4_F32.
// One wave (32 lanes) produces 16 rows of X*w. A-tile: 16x4 f32 per the ISA
// layout (lane m<16: K=0,1 in the two A VGPRs; lane 16+m: K=2,3). B-tile:
// column N=0 carries w, all other columns zero, so accumulator column 0 is
// the running dot product. The zero-mask is applied with a multiply by a
// precomputed 0/1 selector (v_mul_f32), NOT a conditional load — keeps the
// inner loop branch-free and EXEC all-1s (required for WMMA).
// Key packing: score in [0,1) so its IEEE bits are order-preserving;
// key = (score_bits<<32) | ~idx  => sort keys DESCENDING == top_k order.
// ---------------------------------------------------------------------------
__global__ __launch_bounds__(256)
void scores_keys_kernel(const float* __restrict__ X,
                        const float* __restrict__ w,
                        const float* __restrict__ bias,
                        unsigned long long* __restrict__ keys) {
    const int wave = (blockIdx.x * blockDim.x + threadIdx.x) >> 5;
    const int lane = threadIdx.x & 31;
    const int m    = lane & 15;      // row within tile / column N for B
    const int kk   = lane >> 4;      // lane half selects K pair
    const int row0 = wave * 16;

    const float sel = (m == 0) ? 1.0f : 0.0f;   // hoisted B-column mask

    const float* __restrict__ xrow = X + (size_t)(row0 + m) * IN_DIM + kk * 2;
    const float* __restrict__ wp   = w + kk * 2;

    v8f c = {};
#pragma unroll 4
    for (int k = 0; k < IN_DIM; k += 4) {
        const float2 av = *(const float2*)(xrow + k);   // global_load_b64
        const float2 wv = *(const float2*)(wp + k);     // global_load_b64
        v2f a;  a.x = av.x;        a.y = av.y;
        v2f bm; bm.x = wv.x * sel; bm.y = wv.y * sel;   // branch-free mask
        c = __builtin_amdgcn_wmma_f32_16x16x4_f32(
                /*neg_a=*/false, a, /*neg_b=*/false, bm,
                /*c_mod=*/(short)0, c, /*reuse_a=*/false, /*reuse_b=*/false);
    }

    if (m == 0) {   // lanes 0 and 16 hold column 0 of the 16x16 accumulator
        const float bv   = bias[0];
        const int   base = row0 + kk * 8;
#pragma unroll
        for (int i = 0; i < 8; ++i) {
            float dot = c[i] + bv;
            float s   = tanhf(fabsf(dot) * 0.01f);
            unsigned sb = __float_as_uint(s);   // s >= 0 -> bits are monotone
            int ridx = base + i;
            keys[ridx] = ((unsigned long long)sb << 32) |
                         (unsigned)(~(unsigned)ridx);
        }
    }
}

// ---------------------------------------------------------------------------
// Kernel 2: full bitonic sort of all 16384 keys, DESCENDING, inside one
// workgroup. 16384 * 8 B = 128 KB of LDS — fits CDNA5's 320 KB/WGP budget.
// 105 stages, 8 compare-exchanges per thread per stage. Emits idx[0..8191].
// ---------------------------------------------------------------------------
__global__ __launch_bounds__(1024)
void topk_sort_kernel(const unsigned long long* __restrict__ keys_in,
                      int* __restrict__ idx_out) {
    __shared__ unsigned long long k[NTOT];   // 128 KB LDS
    const int tid = threadIdx.x;

    for (int i = tid; i < NTOT; i += 1024) k[i] = keys_in[i];
    __syncthreads();

    for (unsigned size = 2; size <= NTOT; size <<= 1) {
        for (unsigned stride = size >> 1; stride > 0; stride >>= 1) {
#pragma unroll 2
            for (unsigned t = tid; t < NTOT / 2; t += 1024) {
                unsigned i = (t << 1) - (t & (stride - 1));
                unsigned j = i + stride;
                unsigned long long a = k[i];
                unsigned long long b = k[j];
                bool desc = ((i & size) == 0);
                bool sw   = desc ? (a < b) : (a > b);
                if (sw) { k[i] = b; k[j] = a; }
            }
            __syncthreads();
        }
    }

    for (int i = tid; i < NUM_KEEP; i += 1024)
        idx_out[i] = (int)(~(unsigned)k[i]);
}

// ---------------------------------------------------------------------------
// Kernel 3: new_X = X[idx, :]   — coalesced b128 row copies.
// ---------------------------------------------------------------------------
__global__ __launch_bounds__(128)
void gather_x_kernel(const float* __restrict__ X,
                     const int* __restrict__ idx,
                     float* __restrict__ outX) {
    const int row = blockIdx.x;
    const int src = idx[row];          // uniform -> scalar load
    const float4* __restrict__ s =
        (const float4*)(X + (size_t)src * IN_DIM);
    float4* __restrict__ d = (float4*)(outX + (size_t)row * IN_DIM);
    d[threadIdx.x] = s[threadIdx.x];   // 128 threads x 16 B = 2 KB row
}

// ---------------------------------------------------------------------------
// Kernel 4: A_new[i][j] = A[idx[i]][idx[j]]
// One block = 1024 consecutive output columns of one output row.
// Scattered b32 gathers from a single 64 KB source row (prefetched), packed
// into coalesced b128 stores.
// ---------------------------------------------------------------------------
__global__ __launch_bounds__(256)
void gather_a_kernel(const float* __restrict__ A,
                     const int* __restrict__ idx,
                     float* __restrict__ outA) {
    const int i   = blockIdx.y;
    const int src = idx[i];            // uniform
    const float* __restrict__ arow = A + (size_t)src * NTOT;

    // Pre-warm the 64 KB source row: 8 x-blocks * 256 threads * 8-float
    // stride covers all 16384 floats (global_prefetch_b8 per lane).
    __builtin_prefetch(arow + (size_t)blockIdx.x * 2048 + threadIdx.x * 8,
                       0, 3);

    const int jq = blockIdx.x * 256 + threadIdx.x;      // float4 index
    const int4 cols = ((const int4*)idx)[jq];           // 4 column indices

    float4 v;
    v.x = arow[cols.x];
    v.y = arow[cols.y];
    v.z = arow[cols.z];
    v.w = arow[cols.w];
    ((float4*)(outA + (size_t)i * NUM_KEEP))[jq] = v;
}

// ---------------------------------------------------------------------------
// Launch.  d_in: A (NxN f32), X (NxIN_DIM f32), w (IN_DIM f32), b (1 f32).
// d_out (flat f32): A_new (8192*8192) | new_X (8192*512) | idx (8192, i32 bits)
// d_ws: [0,128KB) packed sort keys.
// ---------------------------------------------------------------------------
extern "C" void kernel_launch(void* const* d_in, const int* in_sizes, int n_in,
                              void* d_out, int out_size, void* d_ws, size_t ws_size,
                              hipStream_t stream) {
    const float* A = (const float*)d_in[0];
    const float* X = (const float*)d_in[1];
    const float* w = (const float*)d_in[2];
    const float* b = (const float*)d_in[3];

    float* outA = (float*)d_out;
    float* outX = outA + (size_t)NUM_KEEP * NUM_KEEP;
    int*   outI = (int*)(outX + (size_t)NUM_KEEP * IN_DIM);

    unsigned long long* keys = (unsigned long long*)d_ws;

    // 1) WMMA GEMV -> packed sort keys.  1024 waves = 128 blocks of 256.
    scores_keys_kernel<<<dim3((NTOT / 16) * 32 / 256), dim3(256), 0, stream>>>(
        X, w, b, keys);

    // 2) Single-workgroup bitonic sort in 128 KB LDS -> idx (also an output).
    topk_sort_kernel<<<dim3(1), dim3(1024), 0, stream>>>(keys, outI);

    // 3) Row gather of X.
    gather_x_kernel<<<dim3(NUM_KEEP), dim3(128), 0, stream>>>(X, outI, outX);

    // 4) 2-D gather of A.
    gather_a_kernel<<<dim3(NUM_KEEP / 1024, NUM_KEEP), dim3(256), 0, stream>>>(
        A, outI, outA);
}